// NounPhraseModelV2_81252191306702
// MI455X (gfx1250) — compile-verified
//
#include <hip/hip_runtime.h>
#include <hip/hip_bf16.h>
#include <math.h>

typedef __attribute__((ext_vector_type(2))) float v2f;
typedef __attribute__((ext_vector_type(8))) float v8f;

#define D_HH  256
#define D_XX  64
#define NNN   512
#define LLL   64
#define QQQ   8192
#define NTYPE 16
#define NSEGS 32768
#define NMEMS 262144

#define WMMA_F32(A, B, C) \
    __builtin_amdgcn_wmma_f32_16x16x4_f32(false, (A), false, (B), (short)0, (C), false, false)

// ---------------- init small state ----------------
__global__ void k_init(int* __restrict__ em_ws, unsigned int* __restrict__ acc_cnt) {
    int t = threadIdx.x;
    if (t < NNN) em_ws[t] = 0x7FFFFFFF;   // segment_min identity for int32
    if (t == 0)  *acc_cnt = 0u;
}

// ---------------- segment_sum scatter: x_sum[grp[e]][d] += tok_emb[mem[e]][d] ----------------
__global__ void k_scatter(const float* __restrict__ tok_emb,
                          const int*   __restrict__ mem,
                          const int*   __restrict__ grp,
                          float*       __restrict__ x_sum) {
    unsigned tid = blockIdx.x * blockDim.x + threadIdx.x;
    unsigned e = tid >> 6;
    unsigned d = tid & 63u;
    if (e < NMEMS) {
        int g = grp[e];
        int m = mem[e];
        atomicAdd(&x_sum[(size_t)g * D_XX + d], tok_emb[(size_t)m * D_XX + d]);
    }
}

// ---- kp = h_grp(32768x256) @ Wk.T + bk ; 2x2 register-blocked V_WMMA_F32_16X16X4_F32 ----
__global__ void k_kp(const float* __restrict__ h_grp,
                     const float* __restrict__ Wk,
                     const float* __restrict__ bk,
                     float*       __restrict__ kp) {
    int lane = threadIdx.x;          // 0..31
    int kh   = lane >> 4;            // K-half selector (0: K=k,k+1 ; 1: K=k+2,k+3)
    int mn   = lane & 15;            // A: row m ; B: col n (within 16x16 fragment)
    int mP = blockIdx.x;                                 // 0..1023 : 32-row block
    int nP = blockIdx.y * blockDim.y + threadIdx.y;      // 0..7    : 32-col block
    const float* A0 = h_grp + (size_t)(mP * 32 + mn) * D_HH + 2 * kh;
    const float* A1 = A0 + 16 * D_HH;
    const float* B0 = Wk + (size_t)(nP * 32 + mn) * D_HH + 2 * kh;   // B[k][n]=Wk[n][k]
    const float* B1 = B0 + 16 * D_HH;
    v8f c00 = {}, c01 = {}, c10 = {}, c11 = {};
#pragma unroll 4
    for (int k = 0; k < D_HH; k += 4) {
        v2f a0 = *(const v2f*)(A0 + k);
        v2f a1 = *(const v2f*)(A1 + k);
        v2f b0 = *(const v2f*)(B0 + k);
        v2f b1 = *(const v2f*)(B1 + k);
        c00 = WMMA_F32(a0, b0, c00);
        c01 = WMMA_F32(a0, b1, c01);
        c10 = WMMA_F32(a1, b0, c10);
        c11 = WMMA_F32(a1, b1, c11);
    }
    float bias0 = bk[nP * 32 + mn];
    float bias1 = bk[nP * 32 + 16 + mn];
    int m0 = mP * 32 + kh * 8;       // C/D layout: VGPR r -> M = r + 8*half
    int n0 = nP * 32 + mn;
#pragma unroll
    for (int r = 0; r < 8; ++r) {
        kp[(size_t)(m0 + r)      * D_HH + n0]      = c00[r] + bias0;
        kp[(size_t)(m0 + r)      * D_HH + n0 + 16] = c01[r] + bias1;
        kp[(size_t)(m0 + 16 + r) * D_HH + n0]      = c10[r] + bias0;
        kp[(size_t)(m0 + 16 + r) * D_HH + n0 + 16] = c11[r] + bias1;
    }
}

// ---- qp = concat(h[idx,src], h[idx,dst]) @ Wq.T + bq ; gathered-A, 2x2 blocked WMMA ----
__global__ void k_qp(const float* __restrict__ h_grp,
                     const float* __restrict__ Wq,
                     const float* __restrict__ bq,
                     const int*   __restrict__ idx,
                     const int*   __restrict__ src,
                     const int*   __restrict__ dst,
                     float*       __restrict__ qp) {
    int lane = threadIdx.x;
    int kh   = lane >> 4;
    int mn   = lane & 15;
    int mP = blockIdx.x;                                 // 0..255 : 32 queries
    int nP = blockIdx.y * blockDim.y + threadIdx.y;      // 0..7   : 32 cols
    int q0 = mP * 32 + mn;
    int q1 = q0 + 16;
    int iq0 = idx[q0], iq1 = idx[q1];
    const float* S0 = h_grp + ((size_t)iq0 * LLL + src[q0]) * D_HH + 2 * kh;
    const float* Dd0 = h_grp + ((size_t)iq0 * LLL + dst[q0]) * D_HH + 2 * kh;
    const float* S1 = h_grp + ((size_t)iq1 * LLL + src[q1]) * D_HH + 2 * kh;
    const float* Dd1 = h_grp + ((size_t)iq1 * LLL + dst[q1]) * D_HH + 2 * kh;
    const float* B0 = Wq + (size_t)(nP * 32 + mn) * (2 * D_HH) + 2 * kh;
    const float* B1 = B0 + 16 * (2 * D_HH);
    v8f c00 = {}, c01 = {}, c10 = {}, c11 = {};
#pragma unroll 4
    for (int k = 0; k < D_HH; k += 4) {          // K first half: src rows
        v2f a0 = *(const v2f*)(S0 + k);
        v2f a1 = *(const v2f*)(S1 + k);
        v2f b0 = *(const v2f*)(B0 + k);
        v2f b1 = *(const v2f*)(B1 + k);
        c00 = WMMA_F32(a0, b0, c00);
        c01 = WMMA_F32(a0, b1, c01);
        c10 = WMMA_F32(a1, b0, c10);
        c11 = WMMA_F32(a1, b1, c11);
    }
#pragma unroll 4
    for (int k = 0; k < D_HH; k += 4) {          // K second half: dst rows
        v2f a0 = *(const v2f*)(Dd0 + k);
        v2f a1 = *(const v2f*)(Dd1 + k);
        v2f b0 = *(const v2f*)(B0 + D_HH + k);
        v2f b1 = *(const v2f*)(B1 + D_HH + k);
        c00 = WMMA_F32(a0, b0, c00);
        c01 = WMMA_F32(a0, b1, c01);
        c10 = WMMA_F32(a1, b0, c10);
        c11 = WMMA_F32(a1, b1, c11);
    }
    float bias0 = bq[nP * 32 + mn];
    float bias1 = bq[nP * 32 + 16 + mn];
    int m0 = mP * 32 + kh * 8;
    int n0 = nP * 32 + mn;
#pragma unroll
    for (int r = 0; r < 8; ++r) {
        qp[(size_t)(m0 + r)      * D_HH + n0]      = c00[r] + bias0;
        qp[(size_t)(m0 + r)      * D_HH + n0 + 16] = c01[r] + bias1;
        qp[(size_t)(m0 + 16 + r) * D_HH + n0]      = c10[r] + bias0;
        qp[(size_t)(m0 + 16 + r) * D_HH + n0 + 16] = c11[r] + bias1;
    }
}

// ---------------- per-query attention: scores, masked softmax, attn = a @ x_grp ----------------
__global__ void k_attn(const float* __restrict__ kp,
                       const float* __restrict__ qp,
                       const float* __restrict__ x_sum,
                       const int*   __restrict__ idx,
                       const int*   __restrict__ pos2grp,
                       const int*   __restrict__ msk,
                       float*       __restrict__ attn) {
    __shared__ float qv[D_HH];
    __shared__ float red[LLL];
    __shared__ float av[LLL];
    __shared__ int   p2g[LLL];
    int q  = blockIdx.x;
    int t  = threadIdx.x;   // 0..63 (position l, then output dim d)
    int iq = idx[q];
    for (int j = t; j < D_HH; j += LLL) qv[j] = qp[(size_t)q * D_HH + j];
    p2g[t] = pos2grp[iq * LLL + t];
    int valid = msk[iq * LLL + t];
    __syncthreads();

    const float4* k4 = (const float4*)(kp + ((size_t)iq * LLL + t) * D_HH);
    float s = 0.f;
#pragma unroll 4
    for (int j = 0; j < D_HH / 4; ++j) {
        float4 kk = k4[j];
        s += kk.x * qv[4*j] + kk.y * qv[4*j+1] + kk.z * qv[4*j+2] + kk.w * qv[4*j+3];
    }
    s *= 0.0625f;                       // 1/sqrt(256)
    if (!valid) s = -INFINITY;
    red[t] = s; __syncthreads();
    for (int off = 32; off > 0; off >>= 1) {
        if (t < off) red[t] = fmaxf(red[t], red[t + off]);
        __syncthreads();
    }
    float mx = red[0]; __syncthreads();
    float e = valid ? expf(s - mx) : 0.f;
    red[t] = e; __syncthreads();
    for (int off = 32; off > 0; off >>= 1) {
        if (t < off) red[t] += red[t + off];
        __syncthreads();
    }
    float sm = red[0]; __syncthreads();
    av[t] = e / sm; __syncthreads();

    // attn[q][d] = sum_l a[l] * x_sum[p2g[l]][d], d = t (coalesced over d)
    float acc = 0.f;
    for (int l = 0; l < LLL; ++l)
        acc += av[l] * x_sum[(size_t)p2g[l] * D_XX + t];
    attn[(size_t)q * D_XX + t] = acc;
}

// ---------------- logit = [q_in, attn] @ Wrel.T + brel ; eq metrics ----------------
__global__ void k_logit(const float* __restrict__ h_grp,
                        const float* __restrict__ attn,
                        const float* __restrict__ Wrel,
                        const float* __restrict__ brel,
                        const int*   __restrict__ idx,
                        const int*   __restrict__ src,
                        const int*   __restrict__ dst,
                        const int*   __restrict__ typ,
                        float*       __restrict__ out_logit,
                        int*         __restrict__ em_ws,
                        unsigned int* __restrict__ acc_cnt) {
    __shared__ unsigned int cnt_sh;
    __shared__ unsigned char eqarr[256];
    int tid = threadIdx.x;      // 0..255: 16 queries x 16 types
    int ql  = tid >> 4;
    int t   = tid & 15;
    int q   = blockIdx.x * 16 + ql;
    int iq  = idx[q];
    const float* wr   = Wrel + (size_t)t * (2 * D_HH + D_XX);
    const float* rowS = h_grp + ((size_t)iq * LLL + src[q]) * D_HH;
    const float* rowD = h_grp + ((size_t)iq * LLL + dst[q]) * D_HH;
    float acc = brel[t];
    for (int j = 0; j < D_HH; ++j) acc += rowS[j] * wr[j];
    for (int j = 0; j < D_HH; ++j) acc += rowD[j] * wr[D_HH + j];
    const float* at = attn + (size_t)q * D_XX;
    for (int j = 0; j < D_XX; ++j) acc += at[j] * wr[2 * D_HH + j];
    out_logit[(size_t)q * NTYPE + t] = acc;

    int eq = ((acc > 0.f) == (typ[q * NTYPE + t] > 0)) ? 1 : 0;
    eqarr[tid] = (unsigned char)eq;
    if (tid == 0) cnt_sh = 0u;
    __syncthreads();
    atomicAdd(&cnt_sh, (unsigned)eq);
    if (t == 0) {
        int all = 1;
        for (int j = 0; j < NTYPE; ++j) all &= (int)eqarr[ql * 16 + j];
        atomicMin(&em_ws[iq], all);
    }
    __syncthreads();
    if (tid == 0) atomicAdd(acc_cnt, cnt_sh);
}

// ---------------- finalize: acc, emr, em (as floats) ----------------
__global__ void k_final(const int* __restrict__ em_ws,
                        const unsigned int* __restrict__ acc_cnt,
                        float* __restrict__ d_out) {
    __shared__ float red[512];
    int t = threadIdx.x;     // 0..511
    float e = (float)em_ws[t];
    d_out[131074 + t] = e;                       // em
    red[t] = e; __syncthreads();
    for (int off = 256; off > 0; off >>= 1) {
        if (t < off) red[t] += red[t + off];
        __syncthreads();
    }
    if (t == 0) {
        d_out[131073] = red[0] / 512.f;                               // emr
        d_out[131072] = (float)(*acc_cnt) / (float)(QQQ * NTYPE);     // acc
    }
}

extern "C" void kernel_launch(void* const* d_in, const int* in_sizes, int n_in,
                              void* d_out, int out_size, void* d_ws, size_t ws_size,
                              hipStream_t stream) {
    const float* h_grp   = (const float*)d_in[0];
    const float* tok_emb = (const float*)d_in[1];
    const float* Wq      = (const float*)d_in[2];
    const float* bq      = (const float*)d_in[3];
    const float* Wk      = (const float*)d_in[4];
    const float* bk      = (const float*)d_in[5];
    const float* Wrel    = (const float*)d_in[6];
    const float* brel    = (const float*)d_in[7];
    const int*   mem     = (const int*)d_in[8];
    const int*   grp     = (const int*)d_in[9];
    const int*   pos2grp = (const int*)d_in[10];
    const int*   msk     = (const int*)d_in[11];
    const int*   idx     = (const int*)d_in[12];
    const int*   src     = (const int*)d_in[13];
    const int*   dst     = (const int*)d_in[14];
    const int*   typ     = (const int*)d_in[15];
    float* out = (float*)d_out;

    // workspace carve-up (1 MiB aligned): x_sum 8Mi | kp 32Mi | qp 8Mi | attn 2Mi | em/acc
    char* ws = (char*)d_ws;
    float* x_sum = (float*)(ws);
    float* kp    = (float*)(ws + ((size_t)8  << 20));
    float* qp    = (float*)(ws + ((size_t)40 << 20));
    float* attn  = (float*)(ws + ((size_t)48 << 20));
    int*   em_ws = (int*)  (ws + ((size_t)50 << 20));
    unsigned int* acc_cnt = (unsigned int*)(ws + ((size_t)50 << 20) + 4096);

    hipMemsetAsync(x_sum, 0, (size_t)NSEGS * D_XX * sizeof(float), stream);
    k_init<<<1, 512, 0, stream>>>(em_ws, acc_cnt);
    k_scatter<<<(NMEMS * 64) / 256, 256, 0, stream>>>(tok_emb, mem, grp, x_sum);
    k_kp<<<dim3(NSEGS / 32, 2), dim3(32, 4), 0, stream>>>(h_grp, Wk, bk, kp);
    k_qp<<<dim3(QQQ / 32, 2), dim3(32, 4), 0, stream>>>(h_grp, Wq, bq, idx, src, dst, qp);
    k_attn<<<QQQ, 64, 0, stream>>>(kp, qp, x_sum, idx, pos2grp, msk, attn);
    k_logit<<<QQQ / 16, 256, 0, stream>>>(h_grp, attn, Wrel, brel, idx, src, dst, typ,
                                          out, em_ws, acc_cnt);
    k_final<<<1, 512, 0, stream>>>(em_ws, acc_cnt, out);
}